// PPNet_26912265077423
// MI455X (gfx1250) — compile-verified
//
#include <hip/hip_runtime.h>
#include <hip/hip_bf16.h>
#include <math.h>

typedef __attribute__((ext_vector_type(2))) float v2f;
typedef __attribute__((ext_vector_type(4))) float v4f;
typedef __attribute__((ext_vector_type(8))) float v8f;

#define CC    128          // channels
#define HWPX  196          // 14*14
#define BB    4
#define PP    2000
#define NPIX  (BB*HWPX)    // 784
#define MT    (NPIX/16)    // 49 M-tiles
#define NCLS  200
#define IMG   224
#define IMG2  (IMG*IMG)    // 50176
#define LDA   132          // padded LDS row stride (floats): 132 % 64 == 4 -> conflict free

// ---------------------------------------------------------------------------
// Kernel 1: add_on block.  f = sigmoid(W2 * relu(W1 * fmap + b1) + b2)
// One block = one M-tile of 16 pixels; 8 waves cover the 128 output channels.
// Both GEMMs use V_WMMA_F32_16X16X4_F32 (exact fp32).
// ---------------------------------------------------------------------------
__global__ __launch_bounds__(256) void k_addon(const float* __restrict__ fmap,
                                               const float* __restrict__ w1,
                                               const float* __restrict__ b1,
                                               const float* __restrict__ w2,
                                               const float* __restrict__ b2,
                                               float* __restrict__ f_buf)
{
    __shared__ __align__(16) float Als[16 * LDA];
    __shared__ __align__(16) float Hls[16 * LDA];
    const int t     = threadIdx.x;
    const int mtile = blockIdx.x;

    // Cooperative A-tile load: A[m][k] = fmap[(b*CC+k)*HWPX + s]
    // 16 consecutive threads -> same k, consecutive pixels s -> coalesced.
    for (int i = 0; i < 8; ++i) {
        int idx = i * 256 + t;          // 0..2047
        int m   = idx & 15;
        int k   = idx >> 4;
        int g   = mtile * 16 + m;
        int b   = g / HWPX, s = g % HWPX;
        Als[m * LDA + k] = fmap[(b * CC + k) * HWPX + s];
    }
    __syncthreads();

    const int lane  = t & 31;
    const int wave  = t >> 5;           // 0..7 -> N-tile
    const int row   = lane & 15;        // M for A-frag, N for B-frag/out column
    const int koff  = (lane < 16) ? 0 : 2;
    const int n     = wave * 16 + row;  // output channel this lane owns as column

    // GEMM1: h = relu(A * W1^T + b1).  B[k][n] = w1[n*CC + k]
    v8f acc = {};
    for (int k0 = 0; k0 < CC; k0 += 4) {
        v2f a = *(const v2f*)&Als[row * LDA + k0 + koff];
        v2f bfr = *(const v2f*)&w1[n * CC + k0 + koff];
        acc = __builtin_amdgcn_wmma_f32_16x16x4_f32(false, a, false, bfr,
                                                    (short)0, acc, false, false);
    }
    float bias1 = b1[n];
#pragma unroll
    for (int r = 0; r < 8; ++r) {
        int M = r + ((lane >> 4) << 3);
        Hls[M * LDA + n] = fmaxf(acc[r] + bias1, 0.0f);
    }
    __syncthreads();

    // GEMM2: f = sigmoid(H * W2^T + b2)
    v8f acc2 = {};
    for (int k0 = 0; k0 < CC; k0 += 4) {
        v2f a = *(const v2f*)&Hls[row * LDA + k0 + koff];
        v2f bfr = *(const v2f*)&w2[n * CC + k0 + koff];
        acc2 = __builtin_amdgcn_wmma_f32_16x16x4_f32(false, a, false, bfr,
                                                     (short)0, acc2, false, false);
    }
    float bias2 = b2[n];
#pragma unroll
    for (int r = 0; r < 8; ++r) {
        int M = r + ((lane >> 4) << 3);
        int g = mtile * 16 + M;
        float x  = acc2[r] + bias2;
        float fv = 1.0f / (1.0f + __expf(-x));
        f_buf[g * CC + n] = fv;        // pixel-major: ready as A for the next GEMM
    }
}

// ---------------------------------------------------------------------------
// Kernel 2: squared norms  f2[pix] = sum_c f^2 ;  p2[p] = sum_c proto^2
// ---------------------------------------------------------------------------
__global__ __launch_bounds__(256) void k_sqnorm(const float* __restrict__ f_buf,
                                                const float* __restrict__ protos,
                                                float* __restrict__ f2,
                                                float* __restrict__ p2)
{
    int i = blockIdx.x * blockDim.x + threadIdx.x;
    if (i < NPIX) {
        const v4f* r = (const v4f*)&f_buf[i * CC];
        float s = 0.0f;
        for (int j = 0; j < CC / 4; ++j) {
            v4f v = r[j];
            s += v.x * v.x + v.y * v.y + v.z * v.z + v.w * v.w;
        }
        f2[i] = s;
    } else if (i < NPIX + PP) {
        int p = i - NPIX;
        const v4f* r = (const v4f*)&protos[p * CC];
        float s = 0.0f;
        for (int j = 0; j < CC / 4; ++j) {
            v4f v = r[j];
            s += v.x * v.x + v.y * v.y + v.z * v.z + v.w * v.w;
        }
        p2[p] = s;
    }
}

// ---------------------------------------------------------------------------
// Kernel 3: cross GEMM + L2 distance.
// dist[b][p][s] = sqrt(max(f2 + p2 - 2*<f,proto>, 0) + 1e-12)
// Block = 4 waves sharing one 16-pixel A tile; each wave owns one 16-proto tile.
// ---------------------------------------------------------------------------
__global__ __launch_bounds__(128) void k_dist(const float* __restrict__ f_buf,
                                              const float* __restrict__ protos,
                                              const float* __restrict__ f2,
                                              const float* __restrict__ p2,
                                              float* __restrict__ dist)
{
    __shared__ __align__(16) float Als[16 * LDA];
    const int t     = threadIdx.x;
    const int mtile = blockIdx.x;

    for (int i = 0; i < 16; ++i) {
        int idx = i * 128 + t;          // 0..2047
        int c   = idx & 127;
        int m   = idx >> 7;
        int g   = mtile * 16 + m;
        Als[m * LDA + c] = f_buf[g * CC + c];
    }
    __syncthreads();

    const int lane  = t & 31;
    const int wave  = t >> 5;
    const int ptile = blockIdx.y * 4 + wave;
    if (ptile >= PP / 16) return;       // 125 N-tiles (after the barrier)

    const int row  = lane & 15;
    const int koff = (lane < 16) ? 0 : 2;
    const int p    = ptile * 16 + row;  // this lane's output column (prototype)

    v8f acc = {};
    for (int k0 = 0; k0 < CC; k0 += 4) {
        v2f a = *(const v2f*)&Als[row * LDA + k0 + koff];
        v2f bfr = *(const v2f*)&protos[p * CC + k0 + koff];
        acc = __builtin_amdgcn_wmma_f32_16x16x4_f32(false, a, false, bfr,
                                                    (short)0, acc, false, false);
    }

    float pn = p2[p];
#pragma unroll
    for (int r = 0; r < 8; ++r) {
        int M = r + ((lane >> 4) << 3);
        int g = mtile * 16 + M;
        float d2 = f2[g] + pn - 2.0f * acc[r];
        float d  = sqrtf(fmaxf(d2, 0.0f) + 1e-12f);
        int b = g / HWPX, s = g % HWPX;
        dist[(b * PP + p) * HWPX + s] = d;
    }
}

// ---------------------------------------------------------------------------
// Kernel 4: bilinear 14x14 -> 224x224 (half-pixel centers, edge clamp),
// fused global-min.  One block per (b,p); 1.6 GB streamed out once with
// nontemporal B128 stores; min never re-reads the big buffer.
// ---------------------------------------------------------------------------
__global__ __launch_bounds__(256) void k_upsample(const float* __restrict__ dist,
                                                  float* __restrict__ out_min,
                                                  float* __restrict__ out_up)
{
    __shared__ float tile[HWPX];
    __shared__ int   i0a[IMG];
    __shared__ int   i1a[IMG];
    __shared__ float wa[IMG];
    __shared__ float smin[256];

    const int t  = threadIdx.x;
    const int bp = blockIdx.x;          // 0..7999

    if (t < HWPX) tile[t] = dist[bp * HWPX + t];
    if (t < IMG) {
        float sf = (t + 0.5f) * 0.0625f - 0.5f;   // source coord, scale 1/16
        float fl = floorf(sf);
        int   li = (int)fl;
        wa[t] = sf - fl;
        int lo = li < 0 ? 0 : (li > 13 ? 13 : li);
        int hi = li + 1; hi = hi < 0 ? 0 : (hi > 13 ? 13 : hi);
        i0a[t] = lo;
        i1a[t] = hi;
    }
    __syncthreads();

    float lmin = 3.4e38f;
    float* up = out_up + (size_t)bp * IMG2;
    for (int i = 0; i < 49; ++i) {                // 49*256*4 == 50176 exactly
        int q  = i * 256 + t;                     // float4 index
        int y  = q / 56;
        int x  = (q % 56) * 4;
        int y0 = i0a[y], y1 = i1a[y];
        float wy = wa[y];
        const float* r0 = &tile[y0 * 14];
        const float* r1 = &tile[y1 * 14];
        v4f v;
#pragma unroll
        for (int j = 0; j < 4; ++j) {
            int   xx = x + j;
            int   x0 = i0a[xx], x1 = i1a[xx];
            float wx = wa[xx];
            float a  = r0[x0] + wx * (r0[x1] - r0[x0]);
            float b  = r1[x0] + wx * (r1[x1] - r1[x0]);
            float val = a + wy * (b - a);
            v[j] = val;
            lmin = fminf(lmin, val);
        }
        __builtin_nontemporal_store(v, (v4f*)(up + (size_t)y * IMG + x));
    }

    smin[t] = lmin;
    __syncthreads();
    for (int s = 128; s > 0; s >>= 1) {
        if (t < s) smin[t] = fminf(smin[t], smin[t + s]);
        __syncthreads();
    }
    if (t == 0) out_min[bp] = smin[0];
}

// ---------------------------------------------------------------------------
// Kernel 5: acts = log((md+1)/(md+eps)); logits = acts @ last_w^T
// Activations cached in 32 KB LDS; 800 output dots of length 2000.
// ---------------------------------------------------------------------------
__global__ __launch_bounds__(256) void k_logits(const float* __restrict__ mind,
                                                const float* __restrict__ last_w,
                                                float* __restrict__ logits)
{
    __shared__ __align__(16) float acts[BB * PP];   // 8000 floats
    const int t = threadIdx.x;
    for (int i = t; i < BB * PP; i += 256) {
        float md = mind[i];
        acts[i] = logf((md + 1.0f) / (md + 1e-4f));
    }
    __syncthreads();

    for (int pair = t; pair < BB * NCLS; pair += 256) {
        int b = pair / NCLS, c = pair % NCLS;
        const v4f* wv = (const v4f*)&last_w[c * PP];
        const v4f* av = (const v4f*)&acts[b * PP];
        float s = 0.0f;
        for (int j = 0; j < PP / 4; ++j) {
            v4f w = wv[j];
            v4f a = av[j];
            s += w.x * a.x + w.y * a.y + w.z * a.z + w.w * a.w;
        }
        logits[pair] = s;               // pair == b*NCLS + c
    }
}

// ---------------------------------------------------------------------------
extern "C" void kernel_launch(void* const* d_in, const int* in_sizes, int n_in,
                              void* d_out, int out_size, void* d_ws, size_t ws_size,
                              hipStream_t stream)
{
    const float* fmap   = (const float*)d_in[0];
    const float* w1     = (const float*)d_in[1];
    const float* b1     = (const float*)d_in[2];
    const float* w2     = (const float*)d_in[3];
    const float* b2     = (const float*)d_in[4];
    const float* protos = (const float*)d_in[5];
    const float* last_w = (const float*)d_in[6];

    float* out     = (float*)d_out;
    float* logits  = out;                   // 4*200        = 800
    float* out_min = out + 800;             // 4*2000       = 8000
    float* out_up  = out + 8800;            // 4*2000*224^2 = 401,408,000

    float* ws    = (float*)d_ws;            // ~6.7 MB used
    float* f_buf = ws;                      // 784*128   = 100352
    float* f2    = ws + 100352;             // 784
    float* p2    = ws + 101136;             // 2000
    float* dist  = ws + 103136;             // 4*2000*196 = 1,568,000 (L2-resident)

    k_addon   <<<dim3(MT),          dim3(256), 0, stream>>>(fmap, w1, b1, w2, b2, f_buf);
    k_sqnorm  <<<dim3(11),          dim3(256), 0, stream>>>(f_buf, protos, f2, p2);
    k_dist    <<<dim3(MT, 32),      dim3(128), 0, stream>>>(f_buf, protos, f2, p2, dist);
    k_upsample<<<dim3(BB * PP),     dim3(256), 0, stream>>>(dist, out_min, out_up);
    k_logits  <<<dim3(1),           dim3(256), 0, stream>>>(out_min, last_w, logits);
}